// VQVAE_39556648796227
// MI455X (gfx1250) — compile-verified
//
#include <hip/hip_runtime.h>
#include <math.h>

// ---------------------------------------------------------------------------
// VQ-VAE forward on gfx1250 (MI455X).
// All heavy convs run as implicit GEMM on V_WMMA_F32_16X16X4_F32 (fp32 matrix
// pipe, reference precision). Working set (3 x 33.5MB ping-pong buffers)
// fits in the 192MB L2, so the pipeline is L2-resident after the first load.
// Input pointer order assumes jax pytree flattening (dict keys sorted):
//   params.dec[0]: 0 bt, 1..9 res0(alpha,b1,b2,beta,gamma,mu,var,w1,w2),
//                  10..18 res1, 19 wt
//   params.dec[1]: 20 bt, 21..29 res0, 30..38 res1, 39 wt
//   params.dec_in: 40 alpha, 41 b, 42 w
//   params.embeddings: 43  (D=64, K=512)
//   params.enc[0]: 44 b, 45..53 res0, 54..62 res1, 63 w
//   params.enc[1]: 64 b, 65..73 res0, 74..82 res1, 83 w
//   params.enc_out: 84 alpha, 85 b, 86 w
//   x: 87   (1,128,128,128,1)
// Output: recon (128^3 floats) then perplexity (1 float).
// ---------------------------------------------------------------------------

typedef float v2f __attribute__((ext_vector_type(2)));
typedef float v8f __attribute__((ext_vector_type(8)));

__device__ __forceinline__ v8f wmma_f32_16x16x4(v2f a, v2f b, v8f c) {
  // D = A(16x4) * B(4x16) + C(16x16), fp32 throughout.
  return __builtin_amdgcn_wmma_f32_16x16x4_f32(false, a, false, b, (short)0, c,
                                               false, false);
}

__device__ __forceinline__ v2f ld2(const float* p) {  // 8B-aligned b64 load
  return *(const v2f*)p;
}

// ---------------------------------------------------------------------------
// Generic 3x3x3 SAME conv, NDHWC, implicit GEMM on WMMA.
// MODE 0: out = relu(conv + b)
// MODE 1: out = prelu(conv + b, alpha)
// MODE 2: out = relu(xres + prelu(bnorm(conv + b), alpha))   (res-unit conv2)
// One wave computes a 16-voxel (along W) x 16-Cout tile.
// A-frag: lane holds voxel M = lane%16, channel K = vgpr + 2*(lane/16)
//   -> the two channels per lane are contiguous: single b64 load.
// ---------------------------------------------------------------------------
template <int MODE, int Ci, int Co, int S>
__global__ __launch_bounds__(256) void conv3_wmma(
    const float* __restrict__ in, const float* __restrict__ wgt,
    const float* __restrict__ bias, float* __restrict__ out,
    const float* __restrict__ xres, const float* __restrict__ gamma,
    const float* __restrict__ beta, const float* __restrict__ mu,
    const float* __restrict__ var, const float* __restrict__ alpha) {
  const int WT = S / 16, NT = Co / 16;
  int wave = (blockIdx.x * 256 + threadIdx.x) >> 5;
  int lane = threadIdx.x & 31, half = lane >> 4, lm = lane & 15;
  int nt = wave % NT;
  int vt = wave / NT;
  int w0 = (vt % WT) * 16;
  int h = (vt / WT) % S;
  int d = vt / (WT * S);
  if (d >= S) return;  // wave-uniform
  int n0 = nt * 16;

  v8f acc = {0.f, 0.f, 0.f, 0.f, 0.f, 0.f, 0.f, 0.f};

  for (int dz = 0; dz < 3; ++dz) {
    int z = d + dz - 1;
    bool zv = (unsigned)z < (unsigned)S;
    for (int dy = 0; dy < 3; ++dy) {
      int y = h + dy - 1;
      bool yv = zv && ((unsigned)y < (unsigned)S);
      const float* rowin = in + ((z * S + y) * S) * Ci;
      for (int dx = 0; dx < 3; ++dx) {
        int wx = w0 + lm + dx - 1;
        bool val = yv && ((unsigned)wx < (unsigned)S);
        const float* ain = rowin + wx * Ci + 2 * half;
        const float* bw =
            wgt + (((dz * 3 + dy) * 3 + dx) * Ci + 2 * half) * Co + n0 + lm;
#pragma unroll
        for (int cb = 0; cb < Ci; cb += 4) {
          v2f a = {0.f, 0.f};
          if (val) a = ld2(ain + cb);
          v2f bf;
          bf.x = bw[cb * Co];
          bf.y = bw[(cb + 1) * Co];
          acc = wmma_f32_16x16x4(a, bf, acc);
        }
      }
    }
  }

  int n = n0 + lm;
  float bs = bias[n];
  float bn_scale = 1.f, bn_shift = 0.f, al = 0.f;
  if constexpr (MODE == 2) {
    float g = gamma[n];
    bn_scale = g * rsqrtf(var[n] + 1e-3f);
    bn_shift = beta[n] - mu[n] * bn_scale;
    al = alpha[n];
  }
  if constexpr (MODE == 1) { al = alpha[n]; }

#pragma unroll
  for (int v = 0; v < 8; ++v) {
    int m = v + 8 * half;  // C/D: M = vgpr + 8*(lane/16)
    int off = ((d * S + h) * S + (w0 + m)) * Co + n;
    float x = acc[v] + bs;
    if constexpr (MODE == 0) {
      x = fmaxf(x, 0.f);
    } else if constexpr (MODE == 1) {
      x = (x >= 0.f) ? x : al * x;
    } else {
      x = x * bn_scale + bn_shift;
      x = (x >= 0.f) ? x : al * x;
      x = fmaxf(x + xres[off], 0.f);  // Ci == Co in res-units
    }
    out[off] = x;
  }
}

// ---------------------------------------------------------------------------
// Stride-2, kernel-4 SAME conv (encoder downsample, Cin=32 -> Cout=64), WMMA.
// ---------------------------------------------------------------------------
template <int Ci, int Co, int Si, int So>
__global__ __launch_bounds__(256) void convs2_wmma(
    const float* __restrict__ in, const float* __restrict__ wgt,
    const float* __restrict__ bias, float* __restrict__ out) {
  const int WT = So / 16, NT = Co / 16;
  int wave = (blockIdx.x * 256 + threadIdx.x) >> 5;
  int lane = threadIdx.x & 31, half = lane >> 4, lm = lane & 15;
  int nt = wave % NT;
  int vt = wave / NT;
  int w0 = (vt % WT) * 16;
  int h = (vt / WT) % So;
  int d = vt / (WT * So);
  if (d >= So) return;
  int n0 = nt * 16;

  v8f acc = {0.f, 0.f, 0.f, 0.f, 0.f, 0.f, 0.f, 0.f};
  int wxb = 2 * (w0 + lm) - 1;

  for (int kz = 0; kz < 4; ++kz) {
    int z = 2 * d - 1 + kz;
    bool zv = (unsigned)z < (unsigned)Si;
    for (int ky = 0; ky < 4; ++ky) {
      int y = 2 * h - 1 + ky;
      bool yv = zv && ((unsigned)y < (unsigned)Si);
      const float* rowin = in + ((z * Si + y) * Si) * Ci;
      for (int kx = 0; kx < 4; ++kx) {
        int xw = wxb + kx;
        bool val = yv && ((unsigned)xw < (unsigned)Si);
        const float* ain = rowin + xw * Ci + 2 * half;
        const float* bw =
            wgt + (((kz * 4 + ky) * 4 + kx) * Ci + 2 * half) * Co + n0 + lm;
#pragma unroll
        for (int cb = 0; cb < Ci; cb += 4) {
          v2f a = {0.f, 0.f};
          if (val) a = ld2(ain + cb);
          v2f bf;
          bf.x = bw[cb * Co];
          bf.y = bw[(cb + 1) * Co];
          acc = wmma_f32_16x16x4(a, bf, acc);
        }
      }
    }
  }

  float bs = bias[n0 + lm];
#pragma unroll
  for (int v = 0; v < 8; ++v) {
    int m = v + 8 * half;
    out[((d * So + h) * So + (w0 + m)) * Co + n0 + lm] = fmaxf(acc[v] + bs, 0.f);
  }
}

// ---------------------------------------------------------------------------
// Transposed conv stride-2 kernel-4 (SAME), WMMA via output-parity split:
// each of the 8 parity classes is a dense 2x2x2-tap conv over the input,
// so a 16-voxel M-tile shares weight fragments. out = relu(convT + b).
//   parity p==0: taps k={1,3}, input offset {0,-1}
//   parity p==1: taps k={0,2}, input offset {+1,0}
// ---------------------------------------------------------------------------
template <int Ci, int Co, int I>
__global__ __launch_bounds__(256) void convt_wmma(
    const float* __restrict__ in, const float* __restrict__ wgt,
    const float* __restrict__ bias, float* __restrict__ out) {
  const int O = 2 * I, WT = I / 16, NT = Co / 16;
  int wave = (blockIdx.x * 256 + threadIdx.x) >> 5;
  int lane = threadIdx.x & 31, half = lane >> 4, lm = lane & 15;
  int par = wave & 7;
  int rest = wave >> 3;
  int pw = par & 1, ph = (par >> 1) & 1, pd = (par >> 2) & 1;
  int nt = rest % NT;
  int vt = rest / NT;
  int w0 = (vt % WT) * 16;
  int h = (vt / WT) % I;
  int d = vt / (WT * I);
  if (d >= I) return;
  int n0 = nt * 16;

  const int kd[2] = {pd ? 0 : 1, pd ? 2 : 3}, od[2] = {pd ? 1 : 0, pd ? 0 : -1};
  const int kh[2] = {ph ? 0 : 1, ph ? 2 : 3}, oh[2] = {ph ? 1 : 0, ph ? 0 : -1};
  const int kw[2] = {pw ? 0 : 1, pw ? 2 : 3}, ow[2] = {pw ? 1 : 0, pw ? 0 : -1};

  v8f acc = {0.f, 0.f, 0.f, 0.f, 0.f, 0.f, 0.f, 0.f};

#pragma unroll
  for (int tz = 0; tz < 2; ++tz) {
    int z = d + od[tz];
    bool zv = (unsigned)z < (unsigned)I;
#pragma unroll
    for (int ty = 0; ty < 2; ++ty) {
      int y = h + oh[ty];
      bool yv = zv && ((unsigned)y < (unsigned)I);
      const float* rowin = in + ((z * I + y) * I) * Ci;
#pragma unroll
      for (int tx = 0; tx < 2; ++tx) {
        int xw = w0 + lm + ow[tx];
        bool val = yv && ((unsigned)xw < (unsigned)I);
        const float* ain = rowin + xw * Ci + 2 * half;
        int tap = (kd[tz] * 4 + kh[ty]) * 4 + kw[tx];
        const float* bw = wgt + (tap * Ci + 2 * half) * Co + n0 + lm;
#pragma unroll
        for (int cb = 0; cb < Ci; cb += 4) {
          v2f a = {0.f, 0.f};
          if (val) a = ld2(ain + cb);
          v2f bf;
          bf.x = bw[cb * Co];
          bf.y = bw[(cb + 1) * Co];
          acc = wmma_f32_16x16x4(a, bf, acc);
        }
      }
    }
  }

  float bs = bias[n0 + lm];
#pragma unroll
  for (int v = 0; v < 8; ++v) {
    int m = v + 8 * half;
    int oz = 2 * d + pd, oy = 2 * h + ph, ox = 2 * (w0 + m) + pw;
    out[((oz * O + oy) * O + ox) * Co + n0 + lm] = fmaxf(acc[v] + bs, 0.f);
  }
}

// ---------------------------------------------------------------------------
// Encoder first conv: k4 s2, Cin=1 -> Cout=32, 128^3 -> 64^3, relu. VALU.
// ---------------------------------------------------------------------------
__global__ __launch_bounds__(256) void enc_first(
    const float* __restrict__ x, const float* __restrict__ w,
    const float* __restrict__ b, float* __restrict__ out) {
  int idx = blockIdx.x * 256 + threadIdx.x;
  if (idx >= 64 * 64 * 64 * 32) return;
  int co = idx & 31;
  int v = idx >> 5;
  int ww = v & 63, hh = (v >> 6) & 63, dd = v >> 12;
  float acc = b[co];
  for (int kz = 0; kz < 4; ++kz) {
    int z = 2 * dd - 1 + kz;
    if ((unsigned)z >= 128u) continue;
    for (int ky = 0; ky < 4; ++ky) {
      int y = 2 * hh - 1 + ky;
      if ((unsigned)y >= 128u) continue;
#pragma unroll
      for (int kx = 0; kx < 4; ++kx) {
        int xx = 2 * ww - 1 + kx;
        if ((unsigned)xx >= 128u) continue;
        acc = fmaf(x[(z * 128 + y) * 128 + xx],
                   w[((kz * 4 + ky) * 4 + kx) * 32 + co], acc);
      }
    }
  }
  out[idx] = fmaxf(acc, 0.f);
}

// ---------------------------------------------------------------------------
// Final transposed conv: k4 s2, Cin=32 -> Cout=1, 64^3 -> 128^3, no act. VALU.
// ---------------------------------------------------------------------------
__device__ __forceinline__ void t_taps(int o, int* k, int* i) {
  int p = o & 1, base = o >> 1;
  if (!p) { k[0] = 1; i[0] = base;     k[1] = 3; i[1] = base - 1; }
  else    { k[0] = 0; i[0] = base + 1; k[1] = 2; i[1] = base; }
}

__global__ __launch_bounds__(256) void convt_last(
    const float* __restrict__ in, const float* __restrict__ w,
    const float* __restrict__ b, float* __restrict__ out) {
  int o = blockIdx.x * 256 + threadIdx.x;
  if (o >= 128 * 128 * 128) return;
  int ox = o & 127, oy = (o >> 7) & 127, oz = o >> 14;
  int kz[2], iz[2], ky[2], iy[2], kx[2], ix[2];
  t_taps(oz, kz, iz);
  t_taps(oy, ky, iy);
  t_taps(ox, kx, ix);
  float acc = b[0];
#pragma unroll
  for (int tz = 0; tz < 2; ++tz) {
    if ((unsigned)iz[tz] >= 64u) continue;
#pragma unroll
    for (int ty = 0; ty < 2; ++ty) {
      if ((unsigned)iy[ty] >= 64u) continue;
#pragma unroll
      for (int tx = 0; tx < 2; ++tx) {
        if ((unsigned)ix[tx] >= 64u) continue;
        const float4* ip =
            (const float4*)(in + ((iz[tz] * 64 + iy[ty]) * 64 + ix[tx]) * 32);
        const float4* wp =
            (const float4*)(w + ((kz[tz] * 4 + ky[ty]) * 4 + kx[tx]) * 32);
#pragma unroll
        for (int c = 0; c < 8; ++c) {
          float4 a = ip[c], wv = wp[c];
          acc = fmaf(a.x, wv.x, acc);
          acc = fmaf(a.y, wv.y, acc);
          acc = fmaf(a.z, wv.z, acc);
          acc = fmaf(a.w, wv.w, acc);
        }
      }
    }
  }
  out[o] = acc;
}

// ---------------------------------------------------------------------------
// VQ: code norms, histogram zero, WMMA nearest-code search + gather, perplexity
// ---------------------------------------------------------------------------
__global__ void code_norms(const float* __restrict__ E, float* __restrict__ cn) {
  int n = blockIdx.x * blockDim.x + threadIdx.x;
  if (n >= 512) return;
  float s = 0.f;
  for (int c = 0; c < 64; ++c) { float e = E[c * 512 + n]; s = fmaf(e, e, s); }
  cn[n] = s;
}

__global__ void zero512(float* __restrict__ p) { p[threadIdx.x] = 0.f; }

// One wave handles 16 points. argmin_n (||c_n||^2 - 2 z.c_n) over 512 codes,
// dot products via WMMA fp32 tiles, cross-lane argmin via shfl_xor in 16-lane
// groups (matching the C-fragment layout: N = lane%16, M = vgpr + 8*(lane/16)).
__global__ __launch_bounds__(256) void vq_wmma(
    const float* __restrict__ z, const float* __restrict__ E,
    const float* __restrict__ cnorm, float* __restrict__ q,
    float* __restrict__ hist) {
  int wave = (blockIdx.x * 256 + threadIdx.x) >> 5;
  int lane = threadIdx.x & 31, half = lane >> 4, lm = lane & 15;
  int p0 = wave * 16;
  if (p0 >= 32768) return;

  float bval[8];
  int bidx[8];
#pragma unroll
  for (int v = 0; v < 8; ++v) { bval[v] = 3.4e38f; bidx[v] = 0; }

  const float* arow = z + (p0 + lm) * 64 + 2 * half;
  for (int n0 = 0; n0 < 512; n0 += 16) {
    v8f acc = {0.f, 0.f, 0.f, 0.f, 0.f, 0.f, 0.f, 0.f};
    const float* bwp = E + (2 * half) * 512 + n0 + lm;
#pragma unroll
    for (int cb = 0; cb < 64; cb += 4) {
      v2f a = ld2(arow + cb);
      v2f bf;
      bf.x = bwp[cb * 512];
      bf.y = bwp[(cb + 1) * 512];
      acc = wmma_f32_16x16x4(a, bf, acc);
    }
    float cn = cnorm[n0 + lm];
#pragma unroll
    for (int v = 0; v < 8; ++v) {
      float dr = cn - 2.f * acc[v];
      if (dr < bval[v]) { bval[v] = dr; bidx[v] = n0 + lm; }
    }
  }

#pragma unroll
  for (int v = 0; v < 8; ++v) {
    float bv = bval[v];
    int bi = bidx[v];
    for (int off = 8; off; off >>= 1) {
      float ov = __shfl_xor(bv, off, 16);
      int oi = __shfl_xor(bi, off, 16);
      if (ov < bv || (ov == bv && oi < bi)) { bv = ov; bi = oi; }
    }
    int idxm = __shfl(bi, 0, 16);  // broadcast winner within 16-lane group
    int m = v + 8 * half;          // point index for this (vgpr, half)
    float4 col;                    // gather 4 channels, store as one b128
    col.x = E[(lm * 4 + 0) * 512 + idxm];
    col.y = E[(lm * 4 + 1) * 512 + idxm];
    col.z = E[(lm * 4 + 2) * 512 + idxm];
    col.w = E[(lm * 4 + 3) * 512 + idxm];
    *(float4*)(q + (p0 + m) * 64 + lm * 4) = col;
    if (lm == 0) atomicAdd(hist + idxm, 1.0f);
  }
}

__global__ void perplex(const float* __restrict__ hist, float* __restrict__ out) {
  __shared__ float sm[512];
  int t = threadIdx.x;
  float a = hist[t] * (1.0f / 32768.0f);
  sm[t] = a * logf(a + 1e-10f);
  __syncthreads();
  for (int s = 256; s; s >>= 1) {
    if (t < s) sm[t] += sm[t + s];
    __syncthreads();
  }
  if (t == 0) out[0] = expf(-sm[0]);
}

// ---------------------------------------------------------------------------
// Host-side launch helpers
// ---------------------------------------------------------------------------
template <int MODE, int Ci, int Co, int S>
static void launch_conv3(hipStream_t s, const float* in, const float* w,
                         const float* b, float* out, const float* xres,
                         const float* gamma, const float* beta, const float* mu,
                         const float* var, const float* alpha) {
  int waves = (S / 16) * S * S * (Co / 16);
  conv3_wmma<MODE, Ci, Co, S><<<waves / 8, 256, 0, s>>>(
      in, w, b, out, xres, gamma, beta, mu, var, alpha);
}

// res unit: h = relu(conv1(x)); y = relu(x + prelu(bn(conv2(h))))
// param base ib: +0 alpha, +1 b1, +2 b2, +3 beta, +4 gamma, +5 mu, +6 var,
//                +7 w1, +8 w2
template <int Cc, int S>
static void res_unit(hipStream_t s, const float* const* in, int ib,
                     const float* x, float* h, float* y) {
  launch_conv3<0, Cc, Cc, S>(s, x, in[ib + 7], in[ib + 1], h, nullptr, nullptr,
                             nullptr, nullptr, nullptr, nullptr);
  launch_conv3<2, Cc, Cc, S>(s, h, in[ib + 8], in[ib + 2], y, x, in[ib + 4],
                             in[ib + 3], in[ib + 5], in[ib + 6], in[ib + 0]);
}

extern "C" void kernel_launch(void* const* d_in, const int* in_sizes, int n_in,
                              void* d_out, int out_size, void* d_ws,
                              size_t ws_size, hipStream_t stream) {
  (void)in_sizes; (void)n_in; (void)out_size; (void)ws_size;
  const float* const* in = (const float* const*)d_in;
  float* out = (float*)d_out;
  float* ws = (float*)d_ws;

  const size_t BUF = (size_t)64 * 64 * 64 * 32;  // 8,388,608 floats (33.5 MB)
  float* A = ws;
  float* B = ws + BUF;
  float* C = ws + 2 * BUF;
  float* cnorm = ws + 3 * BUF;
  float* hist = cnorm + 512;

  const float* E = in[43];
  const float* x = in[87];

  // ---- encoder ----
  enc_first<<<(64 * 64 * 64 * 32) / 256, 256, 0, stream>>>(x, in[63], in[44], A);
  res_unit<32, 64>(stream, in, 45, A, B, C);   // enc0 res0 -> C
  res_unit<32, 64>(stream, in, 54, C, B, A);   // enc0 res1 -> A

  {  // enc level 1: 64^3 x 32 -> 32^3 x 64, relu
    int waves = (32 / 16) * 32 * 32 * (64 / 16);
    convs2_wmma<32, 64, 64, 32><<<waves / 8, 256, 0, stream>>>(A, in[83],
                                                               in[64], B);
  }
  res_unit<64, 32>(stream, in, 65, B, C, A);   // enc1 res0 -> A
  res_unit<64, 32>(stream, in, 74, A, C, B);   // enc1 res1 -> B

  // enc_out: prelu(conv3 + b) -> z in A
  launch_conv3<1, 64, 64, 32>(stream, B, in[86], in[85], A, nullptr, nullptr,
                              nullptr, nullptr, nullptr, in[84]);

  // ---- vector quantization ----
  code_norms<<<2, 256, 0, stream>>>(E, cnorm);
  zero512<<<1, 512, 0, stream>>>(hist);
  vq_wmma<<<(32768 / 16) * 32 / 256, 256, 0, stream>>>(A, E, cnorm, B, hist);
  perplex<<<1, 512, 0, stream>>>(hist, out + 128 * 128 * 128);

  // ---- decoder ----
  // dec_in: prelu(conv3 + b): q(B) -> A
  launch_conv3<1, 64, 64, 32>(stream, B, in[42], in[41], A, nullptr, nullptr,
                              nullptr, nullptr, nullptr, in[40]);

  res_unit<64, 32>(stream, in, 1, A, C, B);    // dec0 res0 -> B
  res_unit<64, 32>(stream, in, 10, B, C, A);   // dec0 res1 -> A

  {  // convT level 0: 32^3 x 64 -> 64^3 x 32, relu
    int waves = 8 * (32 / 16) * 32 * 32 * (32 / 16);
    convt_wmma<64, 32, 32><<<waves / 8, 256, 0, stream>>>(A, in[19], in[0], B);
  }

  res_unit<32, 64>(stream, in, 21, B, C, A);   // dec1 res0 -> A
  res_unit<32, 64>(stream, in, 30, A, C, B);   // dec1 res1 -> B

  // final convT: 64^3 x 32 -> 128^3 x 1, no activation, writes recon
  convt_last<<<(128 * 128 * 128) / 256, 256, 0, stream>>>(B, in[39], in[20],
                                                          out);
}